// SAGE_FULL_PYG_38225208934555
// MI455X (gfx1250) — compile-verified
//
#include <hip/hip_runtime.h>

typedef __attribute__((ext_vector_type(2))) float v2f;
typedef __attribute__((ext_vector_type(8))) float v8f;

#define D_FEAT 128  // feature width of both scatter phases (D_IN == D_HID == 128)

// Guaranteed-native f32 atomic add (no-return form -> STOREcnt; s_endpgm waits idle).
__device__ __forceinline__ void atomic_add_f32(float* p, float v) {
    asm volatile("global_atomic_add_f32 %0, %1, off" : : "v"(p), "v"(v) : "memory");
}

// ---------------- zero fill (grid-stride) ----------------
__global__ void zero_f32_kernel(float* __restrict__ p, size_t n) {
    size_t i = (size_t)blockIdx.x * blockDim.x + threadIdx.x;
    size_t stride = (size_t)gridDim.x * blockDim.x;
    for (; i < n; i += stride) p[i] = 0.0f;
}

// ---------------- degree histogram ----------------
__global__ void deg_kernel(const int* __restrict__ dst, float* __restrict__ deg, int E) {
    int t = blockIdx.x * blockDim.x + threadIdx.x;
    if (t < E) atomic_add_f32(&deg[dst[t]], 1.0f);
}

// deg -> deg_inv in place:  d>0 ? 1/d : 0   (d is an integer-valued count)
__global__ void deginv_kernel(float* __restrict__ deg, int N) {
    int t = blockIdx.x * blockDim.x + threadIdx.x;
    if (t < N) {
        float d = deg[t];
        deg[t] = (d > 0.0f) ? (1.0f / d) : 0.0f;
    }
}

// ---------------- weight transpose: WT[n*K + k] = W[k*Ncol + n] ----------------
__global__ void transpose_kernel(const float* __restrict__ W, float* __restrict__ WT,
                                 int K, int Ncol) {
    int t = blockIdx.x * blockDim.x + threadIdx.x;
    if (t < K * Ncol) {
        int k = t / Ncol, n = t % Ncol;
        WT[(size_t)n * K + k] = W[t];
    }
}

// ---------------- edge scatter: agg[dst] += feat[src] ----------------
// One wave per edge (e = t>>5 is wave-uniform -> scalar loads + saddr-form gather).
// Lanes read/write 128 consecutive floats of one row: coalesced 512B gather
// (L2-resident: 51MB << 192MB L2) + lane-contiguous native f32 atomics in L2.
__global__ void scatter_kernel(const float* __restrict__ feat,
                               const int* __restrict__ src,
                               const int* __restrict__ dst,
                               float* __restrict__ agg, int E) {
    long long t = (long long)blockIdx.x * blockDim.x + threadIdx.x;
    int e = __builtin_amdgcn_readfirstlane((int)(t >> 5));
    if (e >= E) return;
    int c = ((int)t & 31) * 4;
    int s = src[e];                       // uniform -> s_load
    int d = dst[e];                       // uniform -> s_load
    const float4 v = *(const float4*)(feat + (size_t)s * D_FEAT + c);
    float* p = agg + (size_t)d * D_FEAT + c;
    atomic_add_f32(p + 0, v.x);
    atomic_add_f32(p + 1, v.y);
    atomic_add_f32(p + 2, v.z);
    atomic_add_f32(p + 3, v.w);
}

// ---------------- fused SAGE GEMM:  out = [relu]( (deg_inv*agg)@Wl + self@Wr + b ) ----
// One wave per 16x16 output tile. K = 128, f32 WMMA 16x16x4.
// A frag (16x4 f32, ISA 7.12.2): lane L holds row M=L&15; lanes 16-31 hold K+2,K+3.
// B frag (4x16) mirrors it; with W pre-transposed (WT[n][k]) each B frag is one b64 load.
// C/D (8 VGPRs): vgpr r, lane L -> M = r + 8*(L>>4), N = L&15.
__global__ void __launch_bounds__(32)
sage_gemm_kernel(const float* __restrict__ agg,
                 const float* __restrict__ self,
                 const float* __restrict__ deg_inv,
                 const float* __restrict__ WlT,   // [ncols x 128] transposed
                 const float* __restrict__ WrT,   // [ncols x 128] transposed
                 const float* __restrict__ bias,
                 float* __restrict__ out,
                 int N, int ncols, int do_relu) {
    const int lane = threadIdx.x;
    const int l15 = lane & 15;
    const int hi = lane >> 4;

    const int rowN = blockIdx.x * 16 + l15;          // logical row for A/deg_inv
    const int rowC = (rowN < N) ? rowN : (N - 1);    // clamp loads, keep EXEC all-1s
    const int bn = blockIdx.y * 16;
    const int col = bn + l15;

    const float dinv = deg_inv[rowC];
    const float* aggRow  = agg  + (size_t)rowC * 128;
    const float* selfRow = self + (size_t)rowC * 128;
    const float* wlCol   = WlT  + (size_t)col * 128;
    const float* wrCol   = WrT  + (size_t)col * 128;

    const float bv = bias[col];
    v8f acc;
#pragma unroll
    for (int r = 0; r < 8; ++r) acc[r] = bv;

#pragma unroll 8
    for (int k = 0; k < 128; k += 4) {
        const int kk = k + 2 * hi;
        v2f a1 = *(const v2f*)(aggRow + kk);
        a1 *= dinv;
        v2f b1 = *(const v2f*)(wlCol + kk);
        v2f a2 = *(const v2f*)(selfRow + kk);
        v2f b2 = *(const v2f*)(wrCol + kk);
        acc = __builtin_amdgcn_wmma_f32_16x16x4_f32(false, a1, false, b1,
                                                    (short)0, acc, false, false);
        acc = __builtin_amdgcn_wmma_f32_16x16x4_f32(false, a2, false, b2,
                                                    (short)0, acc, false, false);
    }

#pragma unroll
    for (int r = 0; r < 8; ++r) {
        int m = blockIdx.x * 16 + r + 8 * hi;
        if (m < N) {
            float v = acc[r];
            if (do_relu) v = fmaxf(v, 0.0f);
            out[(size_t)m * ncols + col] = v;
        }
    }
}

extern "C" void kernel_launch(void* const* d_in, const int* in_sizes, int n_in,
                              void* d_out, int out_size, void* d_ws, size_t ws_size,
                              hipStream_t stream) {
    const float* x   = (const float*)d_in[0];
    const int*   ei  = (const int*)d_in[1];
    const float* Wl0 = (const float*)d_in[2];
    const float* bl0 = (const float*)d_in[3];
    const float* Wr0 = (const float*)d_in[4];
    const float* Wl1 = (const float*)d_in[5];
    const float* bl1 = (const float*)d_in[6];
    const float* Wr1 = (const float*)d_in[7];

    const int D = 128;
    const int HID = 128;
    const int CLS = 64;
    const int N = in_sizes[0] / D;          // 100000
    const int E = in_sizes[1] / 2;          // 1600000
    const int* src = ei;
    const int* dstv = ei + E;

    // workspace layout (floats):
    //   deg/deg_inv [Npad] | agg [N*128] | h [N*128] | WlT0 | WrT0 | WlT1 | WrT1
    float* ws = (float*)d_ws;
    const size_t Npad = ((size_t)N + 3) & ~(size_t)3;
    float* deg  = ws;
    float* agg  = ws + Npad;
    float* h    = agg + (size_t)N * 128;
    float* WlT0 = h + (size_t)N * 128;
    float* WrT0 = WlT0 + (size_t)D * HID;
    float* WlT1 = WrT0 + (size_t)D * HID;
    float* WrT1 = WlT1 + (size_t)HID * CLS;

    const int rowTiles = (N + 15) / 16;
    const long long scatterThreads = (long long)E * 32;
    const int scatterBlocks = (int)((scatterThreads + 255) / 256);

    // ---- prep: zero deg+agg, transpose weights, degrees ----
    zero_f32_kernel<<<2048, 256, 0, stream>>>(deg, Npad + (size_t)N * 128);
    transpose_kernel<<<(D * HID + 255) / 256, 256, 0, stream>>>(Wl0, WlT0, D, HID);
    transpose_kernel<<<(D * HID + 255) / 256, 256, 0, stream>>>(Wr0, WrT0, D, HID);
    transpose_kernel<<<(HID * CLS + 255) / 256, 256, 0, stream>>>(Wl1, WlT1, HID, CLS);
    transpose_kernel<<<(HID * CLS + 255) / 256, 256, 0, stream>>>(Wr1, WrT1, HID, CLS);
    deg_kernel<<<(E + 255) / 256, 256, 0, stream>>>(dstv, deg, E);
    deginv_kernel<<<(N + 255) / 256, 256, 0, stream>>>(deg, N);

    // ---- layer 0: scatter x, then h = relu((deg_inv*agg)@Wl0 + x@Wr0 + b0) ----
    scatter_kernel<<<scatterBlocks, 256, 0, stream>>>(x, src, dstv, agg, E);
    sage_gemm_kernel<<<dim3(rowTiles, HID / 16), 32, 0, stream>>>(
        agg, x, deg, WlT0, WrT0, bl0, h, N, HID, 1);

    // ---- layer 1: re-zero agg, scatter h, out = (deg_inv*agg)@Wl1 + h@Wr1 + b1 ----
    zero_f32_kernel<<<2048, 256, 0, stream>>>(agg, (size_t)N * 128);
    scatter_kernel<<<scatterBlocks, 256, 0, stream>>>(h, src, dstv, agg, E);
    sage_gemm_kernel<<<dim3(rowTiles, CLS / 16), 32, 0, stream>>>(
        agg, h, deg, WlT1, WrT1, bl1, (float*)d_out, N, CLS, 0);
}